// Classifier_5403068859070
// MI455X (gfx1250) — compile-verified
//
#include <hip/hip_runtime.h>
#include <hip/hip_bf16.h>
#include <math.h>

// ---------------------------------------------------------------------------
// GNN classifier for MI455X (gfx1250): fused fp32-WMMA message passing.
//   - V_WMMA_F32_16X16X4_F32, wave32, 16-row tiles per wave
//   - weights pre-transposed & K-padded to 136 for aligned b64 B-fragments
//   - edge MLP fused with gather + atomic scatter-add (L2 resident)
//   - output-tile loops unrolled x2 -> two independent WMMA accumulator
//     chains per wave to hide WMMA latency at low occupancy
// ---------------------------------------------------------------------------

typedef float v2f __attribute__((ext_vector_type(2)));
typedef float v8f __attribute__((ext_vector_type(8)));

#define NN    10000
#define EE    160000
#define EMB   64
#define HIDC  128
#define KPAD  136       // pad 133/130 -> 136 (multiple of 4 for K=4 WMMA steps)
#define GITER 16

__device__ __forceinline__ v8f wmma_f32_k4(v2f a, v2f b, v8f c) {
  // D = A(16x4,f32) * B(4x16,f32) + C(16x16,f32)
  return __builtin_amdgcn_wmma_f32_16x16x4_f32(false, a, false, b, (short)0, c,
                                               false, false);
}

// Permuted e_in layout (so 64-float h segments are float4 aligned in LDS):
//   [0..63]=h[dst] [64..127]=h[src] [128..129]=nf[dst] [130..131]=nf[src]
//   [132]=d [133..135]=pad(0)
__device__ __forceinline__ int edge_src_row(int k) {
  if (k < 64)   return 2 + k;          // h[dst]   (orig rows 2..65)
  if (k < 128)  return 68 + (k - 64);  // h[src]   (orig rows 68..131)
  if (k < 130)  return (k - 128);      // nf[dst]  (orig rows 0..1)
  if (k < 132)  return 66 + (k - 130); // nf[src]  (orig rows 66..67)
  if (k == 132) return 132;            // distance
  return -1;                           // zero pad
}
// Permuted n_in layout: [0..63]=h [64..127]=agg [128..129]=nf [130..135]=pad
__device__ __forceinline__ int node_src_row(int k) {
  if (k < 64)  return k;               // h    (orig rows 0..63)
  if (k < 128) return 66 + (k - 64);   // agg  (orig rows 66..129)
  if (k < 130) return 64 + (k - 128);  // nf   (orig rows 64..65)
  return -1;
}

// wt[n][k] = w[src_row(k)][n], zero for pad rows.  wt: ncols x KPAD
__global__ void pack_w1t_kernel(const float* __restrict__ w,
                                float* __restrict__ wt, int ncols, int mode) {
  int idx = blockIdx.x * blockDim.x + threadIdx.x;
  if (idx >= ncols * KPAD) return;
  int n = idx / KPAD, k = idx - n * KPAD;
  int s = (mode == 1) ? node_src_row(k) : edge_src_row(k);
  wt[idx] = (s >= 0) ? w[s * ncols + n] : 0.0f;
}

// wt[n][k] = w[k][n]  (w: 128x64 row-major -> wt: 64x128)
__global__ void transpose_w2_kernel(const float* __restrict__ w,
                                    float* __restrict__ wt) {
  int idx = blockIdx.x * blockDim.x + threadIdx.x;
  if (idx >= EMB * HIDC) return;
  int n = idx / HIDC, k = idx - n * HIDC;
  wt[idx] = w[k * EMB + n];
}

// h0 = relu(relu(nf @ W1 + b1) @ W2 + b2); tiny (K=2), plain VALU is fine.
__global__ __launch_bounds__(HIDC) void init_embed_kernel(
    const float* __restrict__ nf, const float* __restrict__ w1,
    const float* __restrict__ b1, const float* __restrict__ w2,
    const float* __restrict__ b2, float* __restrict__ h) {
  __shared__ float hid[HIDC];
  const int node = blockIdx.x;
  const int j = threadIdx.x;
  const float f0 = nf[node * 2 + 0], f1 = nf[node * 2 + 1];
  hid[j] = fmaxf(f0 * w1[j] + f1 * w1[HIDC + j] + b1[j], 0.0f);
  __syncthreads();
  if (j < EMB) {
    float s = b2[j];
    #pragma unroll 8
    for (int k = 0; k < HIDC; ++k) s += hid[k] * w2[k * EMB + j];
    h[(size_t)node * EMB + j] = fmaxf(s, 0.0f);
  }
}

// Fused edge MLP: gather e_in tile -> [133->128] -> relu -> [128->64] -> relu
// -> atomicAdd into agg[dst].  One wave per 16 edges, 4 waves/block.
__global__ __launch_bounds__(128) void edge_mlp_kernel(
    const float* __restrict__ nf, const float* __restrict__ h,
    const float* __restrict__ dist, const int* __restrict__ src,
    const int* __restrict__ dst, const float* __restrict__ w1t,
    const float* __restrict__ b1, const float* __restrict__ w2t,
    const float* __restrict__ b2, float* __restrict__ agg) {
  __shared__ __align__(16) float ldsA[4][16][KPAD];
  __shared__ __align__(16) float ldsH[4][16][HIDC];

  const int wave = threadIdx.x >> 5;
  const int lane = threadIdx.x & 31;
  const int lo = lane & 15;        // M row / N col within fragment
  const int hi = lane >> 4;        // K half select
  const int tileBase = (blockIdx.x * 4 + wave) * 16;

  { // gather: 16 edges x 136 features
    const int e = tileBase + lo;
    float* row = ldsA[wave][lo];
    if (hi == 0) {
      const int nd = dst[e];
      const float4* hp = (const float4*)(h + (size_t)nd * EMB);
      float4* rp = (float4*)(row);
      #pragma unroll
      for (int i = 0; i < 16; ++i) rp[i] = hp[i];
      row[128] = nf[nd * 2 + 0];
      row[129] = nf[nd * 2 + 1];
    } else {
      const int ns = src[e];
      const float4* hp = (const float4*)(h + (size_t)ns * EMB);
      float4* rp = (float4*)(row + 64);
      #pragma unroll
      for (int i = 0; i < 16; ++i) rp[i] = hp[i];
      row[130] = nf[ns * 2 + 0];
      row[131] = nf[ns * 2 + 1];
      row[132] = dist[e];
      row[133] = 0.0f; row[134] = 0.0f; row[135] = 0.0f;
    }
  }
  __syncthreads();

  // layer 1: (16 x 136) @ (136 x 128), 8 output tiles of 16.
  // unroll 2 -> two independent accumulator chains per iteration.
  const float* arow1 = ldsA[wave][lo];
  #pragma unroll 2
  for (int nt = 0; nt < HIDC / 16; ++nt) {
    const int n = nt * 16 + lo;
    const float bv = b1[n];
    v8f c;
    #pragma unroll
    for (int i = 0; i < 8; ++i) c[i] = bv;
    const float* brow = w1t + (size_t)n * KPAD;
    #pragma unroll 4
    for (int k0 = 0; k0 < KPAD; k0 += 4) {
      const int kk = k0 + hi * 2;
      v2f a = *(const v2f*)(arow1 + kk);
      v2f b = *(const v2f*)(brow + kk);
      c = wmma_f32_k4(a, b, c);
    }
    #pragma unroll
    for (int r = 0; r < 8; ++r)
      ldsH[wave][r + 8 * hi][nt * 16 + lo] = fmaxf(c[r], 0.0f);
  }
  __syncthreads();

  int dcache[8];
  #pragma unroll
  for (int r = 0; r < 8; ++r) dcache[r] = dst[tileBase + r + 8 * hi];

  // layer 2: (16 x 128) @ (128 x 64), 4 output tiles; scatter-add to agg
  const float* arow2 = ldsH[wave][lo];
  #pragma unroll 2
  for (int nt = 0; nt < EMB / 16; ++nt) {
    const int n = nt * 16 + lo;
    const float bv = b2[n];
    v8f c;
    #pragma unroll
    for (int i = 0; i < 8; ++i) c[i] = bv;
    const float* brow = w2t + (size_t)n * HIDC;
    #pragma unroll 4
    for (int k0 = 0; k0 < HIDC; k0 += 4) {
      const int kk = k0 + hi * 2;
      v2f a = *(const v2f*)(arow2 + kk);
      v2f b = *(const v2f*)(brow + kk);
      c = wmma_f32_k4(a, b, c);
    }
    #pragma unroll
    for (int r = 0; r < 8; ++r) {
      const float v = fmaxf(c[r], 0.0f);
      atomicAdd(&agg[(size_t)dcache[r] * EMB + nt * 16 + lo], v);
    }
  }
}

// Node MLP: n_in = [h | agg | nf | pad] -> [130->128] -> relu -> [128->64].
__global__ __launch_bounds__(128) void node_mlp_kernel(
    const float* __restrict__ nf, const float* __restrict__ h,
    const float* __restrict__ agg, const float* __restrict__ w1t,
    const float* __restrict__ b1, const float* __restrict__ w2t,
    const float* __restrict__ b2, float* __restrict__ hout) {
  __shared__ __align__(16) float ldsA[4][16][KPAD];
  __shared__ __align__(16) float ldsH[4][16][HIDC];
  const int wave = threadIdx.x >> 5;
  const int lane = threadIdx.x & 31;
  const int lo = lane & 15, hi = lane >> 4;
  const int tileBase = (blockIdx.x * 4 + wave) * 16;

  {
    const int node = tileBase + lo;
    float* row = ldsA[wave][lo];
    const float4 z4 = make_float4(0.f, 0.f, 0.f, 0.f);
    if (hi == 0) {
      float4* rp = (float4*)(row);
      if (node < NN) {
        const float4* hp = (const float4*)(h + (size_t)node * EMB);
        #pragma unroll
        for (int i = 0; i < 16; ++i) rp[i] = hp[i];
        row[128] = nf[node * 2 + 0];
        row[129] = nf[node * 2 + 1];
      } else {
        #pragma unroll
        for (int i = 0; i < 16; ++i) rp[i] = z4;
        row[128] = 0.0f; row[129] = 0.0f;
      }
    } else {
      float4* rp = (float4*)(row + 64);
      if (node < NN) {
        const float4* ap = (const float4*)(agg + (size_t)node * EMB);
        #pragma unroll
        for (int i = 0; i < 16; ++i) rp[i] = ap[i];
      } else {
        #pragma unroll
        for (int i = 0; i < 16; ++i) rp[i] = z4;
      }
      #pragma unroll
      for (int k = 130; k < KPAD; ++k) row[k] = 0.0f;
    }
  }
  __syncthreads();

  const float* arow1 = ldsA[wave][lo];
  #pragma unroll 2
  for (int nt = 0; nt < HIDC / 16; ++nt) {
    const int n = nt * 16 + lo;
    const float bv = b1[n];
    v8f c;
    #pragma unroll
    for (int i = 0; i < 8; ++i) c[i] = bv;
    const float* brow = w1t + (size_t)n * KPAD;
    #pragma unroll 4
    for (int k0 = 0; k0 < KPAD; k0 += 4) {
      const int kk = k0 + hi * 2;
      v2f a = *(const v2f*)(arow1 + kk);
      v2f b = *(const v2f*)(brow + kk);
      c = wmma_f32_k4(a, b, c);
    }
    #pragma unroll
    for (int r = 0; r < 8; ++r)
      ldsH[wave][r + 8 * hi][nt * 16 + lo] = fmaxf(c[r], 0.0f);
  }
  __syncthreads();

  const float* arow2 = ldsH[wave][lo];
  #pragma unroll 2
  for (int nt = 0; nt < EMB / 16; ++nt) {
    const int n = nt * 16 + lo;
    const float bv = b2[n];
    v8f c;
    #pragma unroll
    for (int i = 0; i < 8; ++i) c[i] = bv;
    const float* brow = w2t + (size_t)n * HIDC;
    #pragma unroll 4
    for (int k0 = 0; k0 < HIDC; k0 += 4) {
      const int kk = k0 + hi * 2;
      v2f a = *(const v2f*)(arow2 + kk);
      v2f b = *(const v2f*)(brow + kk);
      c = wmma_f32_k4(a, b, c);
    }
    #pragma unroll
    for (int r = 0; r < 8; ++r) {
      const int node = tileBase + r + 8 * hi;
      if (node < NN)
        hout[(size_t)node * EMB + nt * 16 + lo] = fmaxf(c[r], 0.0f);
    }
  }
}

// Final classifier: e_in -> [133->64] relu -> dot(64) -> sigmoid.
__global__ __launch_bounds__(128) void cls_kernel(
    const float* __restrict__ nf, const float* __restrict__ h,
    const float* __restrict__ dist, const int* __restrict__ src,
    const int* __restrict__ dst, const float* __restrict__ w1t,
    const float* __restrict__ b1, const float* __restrict__ w2,
    const float* __restrict__ b2, float* __restrict__ out) {
  __shared__ __align__(16) float ldsA[4][16][KPAD];
  __shared__ __align__(16) float ldsC[4][16][EMB];
  const int wave = threadIdx.x >> 5;
  const int lane = threadIdx.x & 31;
  const int lo = lane & 15, hi = lane >> 4;
  const int tileBase = (blockIdx.x * 4 + wave) * 16;

  {
    const int e = tileBase + lo;
    float* row = ldsA[wave][lo];
    if (hi == 0) {
      const int nd = dst[e];
      const float4* hp = (const float4*)(h + (size_t)nd * EMB);
      float4* rp = (float4*)(row);
      #pragma unroll
      for (int i = 0; i < 16; ++i) rp[i] = hp[i];
      row[128] = nf[nd * 2 + 0];
      row[129] = nf[nd * 2 + 1];
    } else {
      const int ns = src[e];
      const float4* hp = (const float4*)(h + (size_t)ns * EMB);
      float4* rp = (float4*)(row + 64);
      #pragma unroll
      for (int i = 0; i < 16; ++i) rp[i] = hp[i];
      row[130] = nf[ns * 2 + 0];
      row[131] = nf[ns * 2 + 1];
      row[132] = dist[e];
      row[133] = 0.0f; row[134] = 0.0f; row[135] = 0.0f;
    }
  }
  __syncthreads();

  const float* arow = ldsA[wave][lo];
  #pragma unroll 2
  for (int nt = 0; nt < EMB / 16; ++nt) {
    const int n = nt * 16 + lo;
    const float bv = b1[n];
    v8f c;
    #pragma unroll
    for (int i = 0; i < 8; ++i) c[i] = bv;
    const float* brow = w1t + (size_t)n * KPAD;
    #pragma unroll 4
    for (int k0 = 0; k0 < KPAD; k0 += 4) {
      const int kk = k0 + hi * 2;
      v2f a = *(const v2f*)(arow + kk);
      v2f b = *(const v2f*)(brow + kk);
      c = wmma_f32_k4(a, b, c);
    }
    #pragma unroll
    for (int r = 0; r < 8; ++r)
      ldsC[wave][r + 8 * hi][nt * 16 + lo] = fmaxf(c[r], 0.0f);
  }
  __syncthreads();

  if (hi == 0) {
    const int e = tileBase + lo;
    float s = b2[0];
    const float* hv = ldsC[wave][lo];
    #pragma unroll 8
    for (int k = 0; k < EMB; ++k) s += hv[k] * w2[k];
    out[e] = 1.0f / (1.0f + expf(-s));
  }
}

extern "C" void kernel_launch(void* const* d_in, const int* in_sizes, int n_in,
                              void* d_out, int out_size, void* d_ws,
                              size_t ws_size, hipStream_t stream) {
  const float* nf      = (const float*)d_in[0];
  const float* dist    = (const float*)d_in[1];
  const int*   src     = (const int*)d_in[2];
  const int*   dst     = (const int*)d_in[3];
  const float* init_w1 = (const float*)d_in[4];
  const float* init_b1 = (const float*)d_in[5];
  const float* init_w2 = (const float*)d_in[6];
  const float* init_b2 = (const float*)d_in[7];
  const float* edge_w1 = (const float*)d_in[8];
  const float* edge_b1 = (const float*)d_in[9];
  const float* edge_w2 = (const float*)d_in[10];
  const float* edge_b2 = (const float*)d_in[11];
  const float* node_w1 = (const float*)d_in[12];
  const float* node_b1 = (const float*)d_in[13];
  const float* node_w2 = (const float*)d_in[14];
  const float* node_b2 = (const float*)d_in[15];
  const float* cls_w1  = (const float*)d_in[16];
  const float* cls_b1  = (const float*)d_in[17];
  const float* cls_w2  = (const float*)d_in[18];
  const float* cls_b2  = (const float*)d_in[19];

  char* ws = (char*)d_ws;
  size_t off = 0;
  auto wsalloc = [&](size_t bytes) -> void* {
    void* p = ws + off;
    off = (off + bytes + 255) & ~(size_t)255;
    return p;
  };
  float* hA   = (float*)wsalloc((size_t)NN * EMB * sizeof(float));
  float* hB   = (float*)wsalloc((size_t)NN * EMB * sizeof(float));
  float* agg  = (float*)wsalloc((size_t)NN * EMB * sizeof(float));
  float* ew1t = (float*)wsalloc((size_t)HIDC * KPAD * sizeof(float));
  float* nw1t = (float*)wsalloc((size_t)HIDC * KPAD * sizeof(float));
  float* cw1t = (float*)wsalloc((size_t)EMB * KPAD * sizeof(float));
  float* ew2t = (float*)wsalloc((size_t)EMB * HIDC * sizeof(float));
  float* nw2t = (float*)wsalloc((size_t)EMB * HIDC * sizeof(float));

  { // weight packing (runs every launch; deterministic, cheap)
    const int blk = 256;
    const int tot1 = HIDC * KPAD;
    pack_w1t_kernel<<<(tot1 + blk - 1) / blk, blk, 0, stream>>>(edge_w1, ew1t, HIDC, 0);
    pack_w1t_kernel<<<(tot1 + blk - 1) / blk, blk, 0, stream>>>(node_w1, nw1t, HIDC, 1);
    const int totc = EMB * KPAD;
    pack_w1t_kernel<<<(totc + blk - 1) / blk, blk, 0, stream>>>(cls_w1, cw1t, EMB, 2);
    const int tw = EMB * HIDC;
    transpose_w2_kernel<<<(tw + blk - 1) / blk, blk, 0, stream>>>(edge_w2, ew2t);
    transpose_w2_kernel<<<(tw + blk - 1) / blk, blk, 0, stream>>>(node_w2, nw2t);
  }

  init_embed_kernel<<<NN, HIDC, 0, stream>>>(nf, init_w1, init_b1, init_w2,
                                             init_b2, hA);

  float* hcur = hA;
  float* hnext = hB;
  const int edgeBlocks = EE / 64;             // 2500 (exact)
  const int nodeBlocks = (NN + 63) / 64;      // 157
  for (int it = 0; it < GITER; ++it) {
    hipMemsetAsync(agg, 0, (size_t)NN * EMB * sizeof(float), stream);
    edge_mlp_kernel<<<edgeBlocks, 128, 0, stream>>>(
        nf, hcur, dist, src, dst, ew1t, edge_b1, ew2t, edge_b2, agg);
    node_mlp_kernel<<<nodeBlocks, 128, 0, stream>>>(
        nf, hcur, agg, nw1t, node_b1, nw2t, node_b2, hnext);
    float* t = hcur; hcur = hnext; hnext = t;
  }

  cls_kernel<<<edgeBlocks, 128, 0, stream>>>(nf, hcur, dist, src, dst, cw1t,
                                             cls_b1, cls_w2, cls_b2,
                                             (float*)d_out);

  (void)in_sizes; (void)n_in; (void)out_size; (void)ws_size;
}